// PRWKV_7_28793460752445
// MI455X (gfx1250) — compile-verified
//
#include <hip/hip_runtime.h>
#include <math.h>

// ---------------- problem constants ----------------
#define BB  4
#define TT  1024
#define CC  2048
#define HH  32
#define NN  64
#define KVC 512
#define MM  (BB*TT)          // 4096 rows
#define GN_EPS 0.00064f

typedef float v2f __attribute__((ext_vector_type(2)));
typedef float v8f __attribute__((ext_vector_type(8)));

#define AS1 __attribute__((address_space(1)))
#define AS3 __attribute__((address_space(3)))

#if defined(__has_builtin)
# if __has_builtin(__builtin_amdgcn_global_load_async_to_lds_b32) && \
     __has_builtin(__builtin_amdgcn_s_wait_asynccnt)
#  define USE_ASYNC_LDS 1
# endif
#endif

// =====================================================================
// Blocked WMMA f32 GEMM: Out[M,Nc] = X[M,K] @ W^T + bias, W (Nc,K) row-major.
// Wave tile 32x32 (2x2 wmma accs, 1 load per wmma), block = 4 waves -> 64x64.
// grid = (M/64, Nc/64).
// =====================================================================
__global__ __launch_bounds__(128)
void wmma_gemm_f32_blk(const float* __restrict__ X, const float* __restrict__ W,
                       const float* __restrict__ bias, float* __restrict__ Out,
                       int K, int Nc)
{
    const int lane = threadIdx.x & 31;
    const int wave = threadIdx.x >> 5;
    const int half = lane >> 4;
    const int l    = lane & 15;
    const int mw   = wave >> 1, nw = wave & 1;
    const int tileM = blockIdx.x * 64 + mw * 32;
    const int tileN = blockIdx.y * 64 + nw * 32;

    const float* xr0 = X + (size_t)(tileM + l)      * K;
    const float* xr1 = X + (size_t)(tileM + 16 + l) * K;
    const float* wr0 = W + (size_t)(tileN + l)      * K;
    const float* wr1 = W + (size_t)(tileN + 16 + l) * K;

    v8f acc00 = {}, acc01 = {}, acc10 = {}, acc11 = {};
    for (int k0 = 0; k0 < K; k0 += 4) {
        const int ko = k0 + 2*half;
        v2f a0 = *(const v2f*)(xr0 + ko);
        v2f a1 = *(const v2f*)(xr1 + ko);
        v2f b0 = *(const v2f*)(wr0 + ko);
        v2f b1 = *(const v2f*)(wr1 + ko);
        acc00 = __builtin_amdgcn_wmma_f32_16x16x4_f32(false, a0, false, b0, (short)0, acc00, false, false);
        acc01 = __builtin_amdgcn_wmma_f32_16x16x4_f32(false, a0, false, b1, (short)0, acc01, false, false);
        acc10 = __builtin_amdgcn_wmma_f32_16x16x4_f32(false, a1, false, b0, (short)0, acc10, false, false);
        acc11 = __builtin_amdgcn_wmma_f32_16x16x4_f32(false, a1, false, b1, (short)0, acc11, false, false);
    }

    const int col0 = tileN + l, col1 = tileN + 16 + l;
    const float bs0 = bias ? bias[col0] : 0.0f;
    const float bs1 = bias ? bias[col1] : 0.0f;
    #pragma unroll
    for (int j = 0; j < 8; ++j) {
        const int row0 = tileM + j + 8*half;           // C/D: VGPR j -> M=j / j+8
        const int row1 = row0 + 16;
        Out[(size_t)row0 * Nc + col0] = acc00[j] + bs0;
        Out[(size_t)row0 * Nc + col1] = acc01[j] + bs1;
        Out[(size_t)row1 * Nc + col0] = acc10[j] + bs0;
        Out[(size_t)row1 * Nc + col1] = acc11[j] + bs1;
    }
}

// =====================================================================
// Simple WMMA GEMM for the LoRA projections (Nc = 64/32), W stored (K,Nc).
// One wave = one 16x16 tile. TANH applies tanh instead of bias.
// =====================================================================
template<bool TANH>
__global__ __launch_bounds__(128)
void wmma_gemm_f32_small(const float* __restrict__ X, const float* __restrict__ W,
                         float* __restrict__ Out, int K, int Nc)
{
    const int lane = threadIdx.x & 31;
    const int wave = threadIdx.x >> 5;
    const int half = lane >> 4;
    const int l    = lane & 15;
    const int tileM = blockIdx.x * 16;
    const int tileN = wave * 16;
    if (tileN >= Nc) return;             // wave-uniform

    const float* xrow = X + (size_t)(tileM + l) * K;
    const int n = tileN + l;
    v8f acc = {};
    for (int k0 = 0; k0 < K; k0 += 4) {
        v2f a = *(const v2f*)(xrow + k0 + 2*half);
        v2f b;
        b.x = W[(size_t)(k0 + 2*half    ) * Nc + n];
        b.y = W[(size_t)(k0 + 2*half + 1) * Nc + n];
        acc = __builtin_amdgcn_wmma_f32_16x16x4_f32(false, a, false, b, (short)0, acc, false, false);
    }
    #pragma unroll
    for (int j = 0; j < 8; ++j) {
        const int row = tileM + j + 8*half;
        Out[(size_t)row * Nc + n] = TANH ? tanhf(acc[j]) : acc[j];
    }
}

// =====================================================================
// Fused elementwise: LoRA dots (w2/a2/v2), decay, gates, kv-head repeat,
// per-head L2 norm (wave32 shuffle reduce), aa/bb.
// =====================================================================
__global__ __launch_bounds__(256)
void fusion_kernel(const float* __restrict__ klin, const float* __restrict__ vlin,
                   const float* __restrict__ hw,  const float* __restrict__ ha,
                   const float* __restrict__ hv,  const float* __restrict__ v_first,
                   const float* __restrict__ w0,  const float* __restrict__ w2,
                   const float* __restrict__ a0,  const float* __restrict__ a2,
                   const float* __restrict__ v0,  const float* __restrict__ v2,
                   const float* __restrict__ k_k, const float* __restrict__ k_a,
                   float* __restrict__ decay, float* __restrict__ kpost,
                   float* __restrict__ vpost, float* __restrict__ aaO,
                   float* __restrict__ bbO)
{
    const int bt  = blockIdx.x;
    const int tid = threadIdx.x;
    __shared__ float shw[64], sha[64], shv[32];
    if (tid < 64)        shw[tid]       = hw[(size_t)bt*64 + tid];
    else if (tid < 128)  sha[tid - 64]  = ha[(size_t)bt*64 + (tid - 64)];
    else if (tid < 160)  shv[tid - 128] = hv[(size_t)bt*32 + (tid - 128)];
    __syncthreads();

    const int wave = tid >> 5, lane = tid & 31;
    for (int i = 0; i < 4; ++i) {
        const int h = i*8 + wave;
        const int hbase = h * NN;
        float kkp[2], aval[2], kp[2], vp[2], dec[2];
        #pragma unroll
        for (int s = 0; s < 2; ++s) {
            const int n = lane + 32*s;
            const int c = hbase + n;
            const size_t gc = (size_t)bt * CC + c;
            float wsum = w0[c], asum = a0[c], vsum = v0[c];
            for (int j = 0; j < 64; ++j) {
                wsum += shw[j] * w2[(size_t)j*CC + c];
                asum += sha[j] * a2[(size_t)j*CC + c];
            }
            for (int j = 0; j < 32; ++j)
                vsum += shv[j] * v2[(size_t)j*CC + c];
            const float wraw = -log1pf(expf(-wsum)) - 0.6f;   // -softplus(-t)-0.6
            dec[s] = expf(-expf(wraw));
            const float av = 1.f / (1.f + expf(-asum));
            const float vg = 1.f / (1.f + expf(-vsum));
            const int kvc = ((h >> 2) * NN) + n;              // kv-head repeat
            const float kf = klin[(size_t)bt*KVC + kvc];
            const float vf = vlin[(size_t)bt*KVC + kvc];
            kkp[s]  = kf * k_k[c];
            kp[s]   = kf * (1.f + (av - 1.f) * k_a[c]);
            vp[s]   = vf + (v_first[gc] - vf) * vg;
            aval[s] = av;
        }
        float ss = kkp[0]*kkp[0] + kkp[1]*kkp[1];
        for (int off = 16; off; off >>= 1) ss += __shfl_xor(ss, off);
        const float rn = 1.f / fmaxf(sqrtf(ss), 1e-12f);
        #pragma unroll
        for (int s = 0; s < 2; ++s) {
            const int c = hbase + lane + 32*s;
            const size_t gc = (size_t)bt * CC + c;
            const float kkn = kkp[s] * rn;
            decay[gc] = dec[s];
            kpost[gc] = kp[s];
            vpost[gc] = vp[s];
            aaO[gc]   = -kkn;
            bbO[gc]   = kkn * aval[s];
        }
    }
}

// =====================================================================
// Sequential RWKV-7 scan. grid = B*H, block = 64 (one thread per state row),
// S[64] in VGPRs. Per-timestep vectors triple-buffered in LDS; prefetched
// two steps ahead via GLOBAL_LOAD_ASYNC_TO_LDS (ASYNCcnt) when available,
// else software-pipelined register staging.
// =====================================================================
__global__ __launch_bounds__(64)
void scan_kernel(const float* __restrict__ r, const float* __restrict__ dc,
                 const float* __restrict__ k, const float* __restrict__ v,
                 const float* __restrict__ aa, const float* __restrict__ bb,
                 const float* __restrict__ state0, float* __restrict__ y)
{
    const int bh = blockIdx.x;            // b*H + h
    const int b = bh >> 5, h = bh & 31;
    const int i = threadIdx.x;            // state row
    float S[NN];
    const float* st = state0 + ((size_t)bh * NN + i) * NN;
    #pragma unroll
    for (int q = 0; q < NN; ++q) S[q] = st[q];

    __shared__ float buf[3][6*NN];

#ifdef USE_ASYNC_LDS
    // Async global->LDS, 2 steps ahead (12 outstanding b32 loads per wave).
    #define SCAN_ISSUE(t_, slot_)  do {                                          \
        const size_t gb_ = ((size_t)(b*TT + (t_)))*CC + h*NN + i;                \
        float* d_ = &buf[(slot_)][i];                                            \
        __builtin_amdgcn_global_load_async_to_lds_b32((AS1 int*)(r  + gb_), (AS3 int*)(d_        ), 0, 0); \
        __builtin_amdgcn_global_load_async_to_lds_b32((AS1 int*)(dc + gb_), (AS3 int*)(d_ + 1*NN), 0, 0); \
        __builtin_amdgcn_global_load_async_to_lds_b32((AS1 int*)(k  + gb_), (AS3 int*)(d_ + 2*NN), 0, 0); \
        __builtin_amdgcn_global_load_async_to_lds_b32((AS1 int*)(v  + gb_), (AS3 int*)(d_ + 3*NN), 0, 0); \
        __builtin_amdgcn_global_load_async_to_lds_b32((AS1 int*)(aa + gb_), (AS3 int*)(d_ + 4*NN), 0, 0); \
        __builtin_amdgcn_global_load_async_to_lds_b32((AS1 int*)(bb + gb_), (AS3 int*)(d_ + 5*NN), 0, 0); \
    } while (0)

    SCAN_ISSUE(0, 0);
    SCAN_ISSUE(1, 1);
    for (int t = 0; t < TT; ++t) {
        if (t + 2 < TT)      { SCAN_ISSUE(t + 2, (t + 2) % 3);
                               __builtin_amdgcn_s_wait_asynccnt(12); }
        else if (t + 1 < TT) { __builtin_amdgcn_s_wait_asynccnt(6); }
        else                 { __builtin_amdgcn_s_wait_asynccnt(0); }
        __syncthreads();
        const float* rb = &buf[t % 3][0];
        const float* db = rb + NN;  const float* kb = rb + 2*NN;
        const float* vb = rb + 3*NN; const float* ab = rb + 4*NN;
        const float* bp = rb + 5*NN;
        float sa = 0.f;
        #pragma unroll
        for (int q = 0; q < NN; ++q) sa += S[q] * ab[q];
        const float vi = vb[i];
        float yo = 0.f;
        #pragma unroll
        for (int q = 0; q < NN; ++q) {
            const float s2 = S[q]*db[q] + sa*bp[q] + vi*kb[q];
            S[q] = s2;
            yo += s2 * rb[q];
        }
        y[((size_t)(b*TT + t))*CC + h*NN + i] = yo;
        __syncthreads();
    }
    #undef SCAN_ISSUE
#else
    // Fallback: register double-buffer; loads for t+1 issued before compute(t).
    float rg[6];
    {
        const size_t gb = ((size_t)(b*TT))*CC + h*NN + i;
        rg[0]=r[gb]; rg[1]=dc[gb]; rg[2]=k[gb]; rg[3]=v[gb]; rg[4]=aa[gb]; rg[5]=bb[gb];
    }
    for (int t = 0; t < TT; ++t) {
        const int cur = t & 1;
        float* d = &buf[cur][i];
        d[0]=rg[0]; d[1*NN]=rg[1]; d[2*NN]=rg[2]; d[3*NN]=rg[3]; d[4*NN]=rg[4]; d[5*NN]=rg[5];
        __syncthreads();
        if (t + 1 < TT) {
            const size_t gb = ((size_t)(b*TT + t + 1))*CC + h*NN + i;
            rg[0]=r[gb]; rg[1]=dc[gb]; rg[2]=k[gb]; rg[3]=v[gb]; rg[4]=aa[gb]; rg[5]=bb[gb];
        }
        const float* rb = &buf[cur][0];
        const float* db = rb + NN;  const float* kb = rb + 2*NN;
        const float* vb = rb + 3*NN; const float* ab = rb + 4*NN;
        const float* bp = rb + 5*NN;
        float sa = 0.f;
        #pragma unroll
        for (int q = 0; q < NN; ++q) sa += S[q] * ab[q];
        const float vi = vb[i];
        float yo = 0.f;
        #pragma unroll
        for (int q = 0; q < NN; ++q) {
            const float s2 = S[q]*db[q] + sa*bp[q] + vi*kb[q];
            S[q] = s2;
            yo += s2 * rb[q];
        }
        y[((size_t)(b*TT + t))*CC + h*NN + i] = yo;
        __syncthreads();
    }
#endif
}

// =====================================================================
// GroupNorm over (T,N) per (b,h) + bonus term. xx aliases rbuf
// (each lane reads its r before writing the same location).
// =====================================================================
__global__ __launch_bounds__(256)
void gnorm_kernel(const float* __restrict__ y, const float* rbuf,
                  const float* __restrict__ kpost, const float* __restrict__ vpost,
                  const float* __restrict__ r_k, const float* __restrict__ ln_w,
                  const float* __restrict__ ln_b, float* xx)
{
    const int bh = blockIdx.x;
    const int b = bh >> 5, h = bh & 31;
    const int tid = threadIdx.x;
    __shared__ float red[256], red2[256];

    float s = 0.f, s2 = 0.f;
    for (int f = tid; f < TT*NN; f += 256) {
        const int t = f >> 6, n = f & 63;
        const float v = y[((size_t)(b*TT + t))*CC + h*NN + n];
        s += v; s2 += v*v;
    }
    red[tid] = s; red2[tid] = s2;
    __syncthreads();
    for (int off = 128; off; off >>= 1) {
        if (tid < off) { red[tid] += red[tid+off]; red2[tid] += red2[tid+off]; }
        __syncthreads();
    }
    const float mean = red[0]  * (1.f / 65536.f);
    const float var  = red2[0] * (1.f / 65536.f) - mean*mean;
    const float rstd = rsqrtf(var + GN_EPS);

    const int wave = tid >> 5, lane = tid & 31;
    const float lw0 = ln_w[h*NN + lane],      lb0 = ln_b[h*NN + lane];
    const float lw1 = ln_w[h*NN + lane + 32], lb1 = ln_b[h*NN + lane + 32];
    const float rk0 = r_k[h*NN + lane],       rk1 = r_k[h*NN + lane + 32];
    for (int t = wave; t < TT; t += 8) {
        const size_t base = ((size_t)(b*TT + t))*CC + h*NN;
        const float r0 = rbuf[base + lane],      r1 = rbuf[base + lane + 32];
        const float k0 = kpost[base + lane],     k1 = kpost[base + lane + 32];
        float d = r0*k0*rk0 + r1*k1*rk1;
        for (int off = 16; off; off >>= 1) d += __shfl_xor(d, off);
        const float y0 = y[base + lane], y1 = y[base + lane + 32];
        const float o0 = (y0 - mean)*rstd*lw0 + lb0 + d*vpost[base + lane];
        const float o1 = (y1 - mean)*rstd*lw1 + lb1 + d*vpost[base + lane + 32];
        xx[base + lane]      = o0;
        xx[base + lane + 32] = o1;
    }
}

// =====================================================================
extern "C" void kernel_launch(void* const* d_in, const int* in_sizes, int n_in,
                              void* d_out, int out_size, void* d_ws, size_t ws_size,
                              hipStream_t stream) {
    const float* x       = (const float*)d_in[0];
    const float* v_first = (const float*)d_in[1];
    const float* state0  = (const float*)d_in[2];
    const float* Rw      = (const float*)d_in[3];
    const float* R_bias  = (const float*)d_in[4];
    const float* Kw      = (const float*)d_in[5];
    const float* K_bias  = (const float*)d_in[6];
    const float* Vw      = (const float*)d_in[7];
    const float* V_bias  = (const float*)d_in[8];
    const float* Ow      = (const float*)d_in[9];
    const float* O_bias  = (const float*)d_in[10];
    const float* w0      = (const float*)d_in[11];
    const float* w1      = (const float*)d_in[12];
    const float* w2      = (const float*)d_in[13];
    const float* a0      = (const float*)d_in[14];
    const float* a1      = (const float*)d_in[15];
    const float* a2      = (const float*)d_in[16];
    const float* v0      = (const float*)d_in[17];
    const float* v1      = (const float*)d_in[18];
    const float* v2      = (const float*)d_in[19];
    const float* k_k     = (const float*)d_in[20];
    const float* k_a     = (const float*)d_in[21];
    const float* r_k     = (const float*)d_in[22];
    const float* ln_w    = (const float*)d_in[23];
    const float* ln_b    = (const float*)d_in[24];
    float* out = (float*)d_out;

    // workspace layout (floats)
    float* ws = (float*)d_ws;
    float* r_buf  = ws;                                // M*C   (later reused as xx)
    float* klin   = r_buf  + (size_t)MM*CC;            // M*512
    float* vlin   = klin   + (size_t)MM*KVC;           // M*512
    float* hw     = vlin   + (size_t)MM*KVC;           // M*64
    float* ha     = hw     + (size_t)MM*64;            // M*64
    float* hv     = ha     + (size_t)MM*64;            // M*32
    float* decayb = hv     + (size_t)MM*32;            // M*C
    float* kpost  = decayb + (size_t)MM*CC;            // M*C
    float* vpost  = kpost  + (size_t)MM*CC;            // M*C
    float* aab    = vpost  + (size_t)MM*CC;            // M*C
    float* bbb    = aab    + (size_t)MM*CC;            // M*C
    float* ybuf   = bbb    + (size_t)MM*CC;            // M*C
    float* xx     = r_buf;                             // alias: r dead after bonus read

    const dim3 blk(128);
    // Step 1: projections (blocked WMMA for the big GEMMs)
    wmma_gemm_f32_blk<<<dim3(MM/64, CC/64),  blk, 0, stream>>>(x, Rw, R_bias, r_buf, CC, CC);
    wmma_gemm_f32_blk<<<dim3(MM/64, KVC/64), blk, 0, stream>>>(x, Kw, K_bias, klin,  CC, KVC);
    wmma_gemm_f32_blk<<<dim3(MM/64, KVC/64), blk, 0, stream>>>(x, Vw, V_bias, vlin,  CC, KVC);
    wmma_gemm_f32_small<true ><<<dim3(MM/16), blk, 0, stream>>>(x, w1, hw, CC, 64);
    wmma_gemm_f32_small<false><<<dim3(MM/16), blk, 0, stream>>>(x, a1, ha, CC, 64);
    wmma_gemm_f32_small<false><<<dim3(MM/16), blk, 0, stream>>>(x, v1, hv, CC, 32);

    // fused elementwise prep
    fusion_kernel<<<dim3(MM), dim3(256), 0, stream>>>(
        klin, vlin, hw, ha, hv, v_first, w0, w2, a0, a2, v0, v2, k_k, k_a,
        decayb, kpost, vpost, aab, bbb);

    // Step 2: sequential scan
    scan_kernel<<<dim3(BB*HH), dim3(64), 0, stream>>>(
        r_buf, decayb, kpost, vpost, aab, bbb, state0, ybuf);

    // Step 3: group norm + bonus
    gnorm_kernel<<<dim3(BB*HH), dim3(256), 0, stream>>>(
        ybuf, r_buf, kpost, vpost, r_k, ln_w, ln_b, xx);

    // output projection
    wmma_gemm_f32_blk<<<dim3(MM/64, CC/64), blk, 0, stream>>>(xx, Ow, O_bias, out, CC, CC);
    (void)in_sizes; (void)n_in; (void)out_size; (void)ws_size;
}